// QDPPMixer_17952963297357
// MI455X (gfx1250) — compile-verified
//
#include <hip/hip_runtime.h>

typedef __attribute__((ext_vector_type(2))) float v2f;
typedef __attribute__((ext_vector_type(8))) float v8f;

#define NA 8          // agents
#define NACT 10       // actions
#define EMB 32        // embed dim
#define BIAS 1000     // STATE_NUM * N_ACTIONS

// ds_swizzle group-of-32 mode: new_lane = ((lane & AND) | OR) ^ XOR
// pattern = (XOR<<10) | (OR<<5) | AND    (offset[15]=0)
#define SWZ_BCAST8(x, K) \
  __int_as_float(__builtin_amdgcn_ds_swizzle(__float_as_int(x), ((K) << 5) | 24))
#define SWZ_XOR(x, M) \
  __int_as_float(__builtin_amdgcn_ds_swizzle(__float_as_int(x), ((M) << 10) | 31))

// One wave handles 2 batch elements: their 8x8 Gram matrices are the two
// diagonal blocks of a single 16x16 f32 WMMA accumulation (K=32 via 8 chained
// V_WMMA_F32_16X16X4_F32). The f32 16x4 A layout (lanes 0-15: K=0,1; lanes
// 16-31: K=2,3) mirrors the 4x16 B layout, so for a Gram matrix the SAME
// register pair serves as both SRC0 and SRC1. Elem0's 8x8 block lands
// column-wise in lanes 0-7, elem1's in lanes 24-31; both LU factorizations run
// in place in those 8-lane groups (lanes 8-23 compute discarded garbage).
__global__ __launch_bounds__(256) void qdpp_mix_kernel(
    const float* __restrict__ agent_qs,
    const int*   __restrict__ states,
    const int*   __restrict__ actions,
    const float* __restrict__ noise,
    const float* __restrict__ Bw,
    float*       __restrict__ out,
    int npairs)
{
  const int lane = threadIdx.x & 31;
  const int wave = threadIdx.x >> 5;
  int pair = blockIdx.x * 8 + wave;
  if (pair >= npairs) pair = npairs - 1;  // clamp (not predicate): EXEC stays all-ones for WMMA

  const int hi = lane >> 4;        // half-wave: k mod 4 in {0,1} vs {2,3}
  const int r  = lane & 15;        // matrix row 0..15 (0-7 = elem0, 8-15 = elem1)
  const int e  = pair * 2 + (r >> 3);
  const int ag = r & 7;

  // embedding gather index: state*10 + action + 1000*agent
  const int st  = states[e * NA + ag];
  const int ac  = actions[e * NA + ag];
  const int row = st * NACT + ac + BIAS * ag;
  const float* rp = Bw + (size_t)row * EMB + 2 * hi;

  // each lane loads its 16 of the 32 row values: k = 4s + 2*hi + {0,1}
  float v[16];
  #pragma unroll
  for (int s = 0; s < 8; ++s) {
    const float2 t = *(const float2*)(rp + 4 * s);
    v[2 * s]     = t.x;
    v[2 * s + 1] = t.y;
  }

  // L2-normalize the row: lane L and L+16 hold complementary halves
  float ssq = 0.f;
  #pragma unroll
  for (int j = 0; j < 16; ++j) ssq = fmaf(v[j], v[j], ssq);
  ssq += SWZ_XOR(ssq, 16);
  const float rn = __builtin_amdgcn_rsqf(fmaxf(ssq, 1e-24f));  // 1/max(||b||,1e-12)
  #pragma unroll
  for (int j = 0; j < 16; ++j) v[j] *= rn;

  // clip(agent_qs) and sum over 8 agents; the xor tree stays inside each
  // 8-lane group, so lanes 0-7 end with e0's sum and lanes 24-31 with e1's.
  float qv = agent_qs[e * NA + ag];
  qv = fminf(fmaxf(qv, -10.f), 10.f);
  qv += SWZ_XOR(qv, 1);
  qv += SWZ_XOR(qv, 2);
  qv += SWZ_XOR(qv, 4);

  // Gram matrix: 8 chained f32 WMMAs, A == B (same registers)
  v8f c = {};
  #pragma unroll
  for (int s = 0; s < 8; ++s) {
    v2f a;
    a[0] = v[2 * s];
    a[1] = v[2 * s + 1];
    c = __builtin_amdgcn_wmma_f32_16x16x4_f32(
        false, a, false, a, (short)0, c, false, false);
  }

  // C layout: lane L, vgpr i -> G[i + 8*(L>=16)][L%16].
  // lanes 0-7:   e0 block, column = lane,    rows in vgprs 0-7
  // lanes 24-31: e1 block, column = lane-24, rows in vgprs 0-7
  // diag(noise)*0.1, relu, +eps   (column index within block == ag)
  const float nv = 0.1f * noise[e * NA + ag];
  float m[8];
  #pragma unroll
  for (int i = 0; i < 8; ++i) {
    const float x = c[i] + (i == ag ? nv : 0.f);
    m[i] = fmaxf(x, 0.f) + 1e-8f;
  }

  // lockstep column-parallel LU (no pivoting) within each 8-lane group;
  // broadcasts use immediate-pattern ds_swizzle (no index VGPR math)
  float det = 1.f;
  #define LU_STEP(K)                                          \
  {                                                           \
    const float akk = SWZ_BCAST8(m[K], K);  /* pivot u_kk */  \
    det *= akk;                                               \
    const float ia = __builtin_amdgcn_rcpf(akk);              \
    _Pragma("unroll")                                         \
    for (int i = (K) + 1; i < 8; ++i) {                       \
      const float lik = SWZ_BCAST8(m[i], K) * ia;             \
      m[i] = fmaf(-lik, m[K], m[i]);                          \
    }                                                         \
  }
  LU_STEP(0) LU_STEP(1) LU_STEP(2) LU_STEP(3)
  LU_STEP(4) LU_STEP(5) LU_STEP(6) LU_STEP(7)
  #undef LU_STEP

  if (lane == 0 || lane == 24) {
    // ln(x) = log2(x) * ln(2)
    const float logdet = 0.69314718056f * __builtin_amdgcn_logf(det + 1e-8f);
    out[e] = qv + logdet;
  }
}

extern "C" void kernel_launch(void* const* d_in, const int* in_sizes, int n_in,
                              void* d_out, int out_size, void* d_ws, size_t ws_size,
                              hipStream_t stream) {
  const float* agent_qs = (const float*)d_in[0];
  const int*   states   = (const int*)d_in[1];
  const int*   actions  = (const int*)d_in[2];
  const float* noisep   = (const float*)d_in[3];
  const float* Bw       = (const float*)d_in[4];
  float* out = (float*)d_out;

  const int nelem  = in_sizes[0] / NA;      // 4096*64 = 262144
  const int npairs = (nelem + 1) / 2;       // 131072
  const int blocks = (npairs + 7) / 8;      // 8 waves (2 elems each) per 256-thread block

  qdpp_mix_kernel<<<blocks, 256, 0, stream>>>(agent_qs, states, actions,
                                              noisep, Bw, out, npairs);
}